// voxelization_29815662969438
// MI455X (gfx1250) — compile-verified
//
#include <hip/hip_runtime.h>
#include <math.h>

// ---------------------------------------------------------------------------
// Voxelization + multi-scale unique for MI455X (gfx1250, wave32).
//
// Bounded 32-bit key space per scale -> presence bitmap + two-level popcount
// prefix gives O(1) sorted-unique rank:
//   rank(key) = wordPrefix[key>>5] + blockSums[(key>>5)>>8]
//             + popc(bitmap[key>>5] & ((1<<bit)-1))
// Bitmap order == ascending key order, reproducing
// jnp.unique(key, return_inverse=True, size=N, fill_value=-1) exactly
// (padding rows handled by pre-filling coors with the key==-1 decode).
// ---------------------------------------------------------------------------

#define TPB 256

// Per-point: polar convert, quantize, flag key bit, emit bxyz row, cache key.
__global__ void vox_point_flag_kernel(const float* __restrict__ pts,
                                      const int* __restrict__ batch,
                                      unsigned* __restrict__ bitmap,
                                      unsigned* __restrict__ keyCache,
                                      float4* __restrict__ bxyz,
                                      float ss0, float ss1, float ss2,
                                      int sx, int sy, int sz, int n) {
    int i = blockIdx.x * TPB + threadIdx.x;
    if (i >= n) return;
    float x = pts[(size_t)i * 4 + 0];
    float y = pts[(size_t)i * 4 + 1];
    float z = pts[(size_t)i * 4 + 2];

    float rho = sqrtf(x * x + y * y);
    float phi = atan2f(y, x);

    // POL_MIN / (POL_MAX - POL_MIN) in exact fp32 (constant-folded).
    const float minb0 = 0.0f;
    const float minb1 = -3.14159265f;                 // rounds to -pi(f32)
    const float minb2 = -4.0f;
    const float crop0 = 50.0f - 0.0f;
    const float crop1 = 3.14159265f - (-3.14159265f); // 2*pi(f32), exact
    const float crop2 = 2.0f - (-4.0f);

    // idx = trunc( ss * (pol - minb) / crop ), all components >= 0
    int ix = (int)((ss0 * (rho - minb0)) / crop0);
    int iy = (int)((ss1 * (phi - minb1)) / crop1);
    int iz = (int)((ss2 * (z   - minb2)) / crop2);
    int b  = batch[i];

    unsigned key = ((((unsigned)b * (unsigned)sx + (unsigned)ix) * (unsigned)sy
                     + (unsigned)iy) * (unsigned)sz) + (unsigned)iz;

    atomicOr(&bitmap[key >> 5], 1u << (key & 31u));
    keyCache[i] = key;

    bxyz[i] = make_float4((float)b, (float)ix, (float)iy, (float)iz);
}

// Per-256-word block: popcount each bitmap word, exclusive scan within block.
__global__ void vox_scan_words_kernel(const unsigned* __restrict__ bitmap,
                                      unsigned* __restrict__ wordPrefix,
                                      unsigned* __restrict__ blockSums,
                                      int nwords) {
    __shared__ unsigned smem[TPB];
    int tid = threadIdx.x;
    int i = blockIdx.x * TPB + tid;
    unsigned v = (i < nwords) ? (unsigned)__popc(bitmap[i]) : 0u;
    smem[tid] = v;
    __syncthreads();
    #pragma unroll
    for (int off = 1; off < TPB; off <<= 1) {
        unsigned t = (tid >= off) ? smem[tid - off] : 0u;
        __syncthreads();
        smem[tid] += t;
        __syncthreads();
    }
    unsigned incl = smem[tid];
    if (i < nwords) wordPrefix[i] = incl - v;          // exclusive in block
    if (tid == TPB - 1) blockSums[blockIdx.x] = incl;  // block total
}

// Single-block exclusive scan of blockSums (in place), chunked.
__global__ void vox_scan_blocksums_kernel(unsigned* __restrict__ blockSums, int nblk) {
    __shared__ unsigned smem[TPB];
    __shared__ unsigned runningOffset;
    int tid = threadIdx.x;
    if (tid == 0) runningOffset = 0u;
    __syncthreads();
    for (int base = 0; base < nblk; base += TPB) {
        int i = base + tid;
        unsigned v = (i < nblk) ? blockSums[i] : 0u;
        smem[tid] = v;
        __syncthreads();
        for (int off = 1; off < TPB; off <<= 1) {
            unsigned t = (tid >= off) ? smem[tid - off] : 0u;
            __syncthreads();
            smem[tid] += t;
            __syncthreads();
        }
        unsigned incl = smem[tid];
        unsigned chunkTotal = smem[TPB - 1];
        unsigned ro = runningOffset;
        if (i < nblk) blockSums[i] = (incl - v) + ro;
        __syncthreads();
        if (tid == 0) runningOffset = ro + chunkTotal;
        __syncthreads();
    }
}

// Pre-fill coors with the unq_key == -1 padding row: [-1, sz-1, sy-1, sx-1]
// (numpy floor-div/mod semantics for a negative key).
__global__ void vox_fill_coors_kernel(float4* __restrict__ coors,
                                      int sx, int sy, int sz, int n) {
    int i = blockIdx.x * TPB + threadIdx.x;
    if (i >= n) return;
    coors[i] = make_float4(-1.0f, (float)(sz - 1), (float)(sy - 1),
                           (float)(sx - 1));
}

// One thread per key in keyspace: if present, compute sorted rank via the
// two-level prefix and emit decoded coors row [b, z, y, x] at that rank.
__global__ void vox_emit_unique_kernel(const unsigned* __restrict__ bitmap,
                                       const unsigned* __restrict__ wordPrefix,
                                       const unsigned* __restrict__ blockSums,
                                       float4* __restrict__ coors,
                                       int sx, int sy, int sz, unsigned K) {
    unsigned key = blockIdx.x * TPB + threadIdx.x;
    if (key >= K) return;
    unsigned word = key >> 5;
    unsigned w = bitmap[word];
    unsigned bit = key & 31u;
    if (!((w >> bit) & 1u)) return;
    unsigned rank = wordPrefix[word] + blockSums[word >> 8]
                  + (unsigned)__popc(w & ((1u << bit) - 1u));
    unsigned t = key;
    unsigned uz = t % (unsigned)sz; t /= (unsigned)sz;
    unsigned uy = t % (unsigned)sy; t /= (unsigned)sy;
    unsigned ux = t % (unsigned)sx;
    unsigned ub = t / (unsigned)sx;
    coors[rank] = make_float4((float)ub, (float)uz, (float)uy, (float)ux);
}

// Per-point inverse index via O(1) rank lookup (two-level prefix).
__global__ void vox_inv_kernel(const unsigned* __restrict__ bitmap,
                               const unsigned* __restrict__ wordPrefix,
                               const unsigned* __restrict__ blockSums,
                               const unsigned* __restrict__ keyCache,
                               float* __restrict__ inv, int n) {
    int i = blockIdx.x * TPB + threadIdx.x;
    if (i >= n) return;
    unsigned key = keyCache[i];
    unsigned word = key >> 5;
    unsigned w = bitmap[word];
    unsigned bit = key & 31u;
    unsigned rank = wordPrefix[word] + blockSums[word >> 8]
                  + (unsigned)__popc(w & ((1u << bit) - 1u));
    inv[i] = (float)rank;
}

extern "C" void kernel_launch(void* const* d_in, const int* in_sizes, int n_in,
                              void* d_out, int out_size, void* d_ws, size_t ws_size,
                              hipStream_t stream) {
    const float* pts   = (const float*)d_in[0];  // (N,4) float32
    const int*   batch = (const int*)d_in[1];    // (N,)  int
    float* out = (float*)d_out;
    const int n = in_sizes[1];                   // N points

    // Workspace layout (u32 units). Max key space (scale=1): 4*481*361*33 =
    // 22,920,612 keys -> 716,270 bitmap words; round to 716,288 (mult of 256).
    const int MAXW = 716288;
    unsigned* ws         = (unsigned*)d_ws;
    unsigned* bitmap     = ws;                            // MAXW words
    unsigned* wordPrefix = ws + (size_t)MAXW;             // MAXW words
    unsigned* blockSums  = ws + 2 * (size_t)MAXW;         // 4096 words
    unsigned* keyCache   = ws + 2 * (size_t)MAXW + 4096;  // N words

    const int scales[5] = {2, 4, 8, 16, 1};
    const int pgrid = (n + TPB - 1) / TPB;
    size_t base = 0;

    for (int s = 0; s < 5; ++s) {
        const int scale = scales[s];
        const float ss0 = ceilf(480.0f / (float)scale);
        const float ss1 = ceilf(360.0f / (float)scale);
        const float ss2 = ceilf(32.0f  / (float)scale);
        const int sx = (int)ss0 + 1;
        const int sy = (int)ss1 + 1;
        const int sz = (int)ss2 + 1;
        const unsigned K = (unsigned)(4 * sx * sy * sz);  // BS=4
        const int nwords = (int)((K + 31u) / 32u);
        const int nblk = (nwords + TPB - 1) / TPB;

        float4* bxyz  = (float4*)(out + base);            // (N,4)
        float*  inv   = out + base + (size_t)n * 4;       // (N,)
        float4* coors = (float4*)(out + base + (size_t)n * 5); // (N,4)
        base += (size_t)n * 9;

        hipMemsetAsync(bitmap, 0, (size_t)nwords * sizeof(unsigned), stream);

        vox_point_flag_kernel<<<pgrid, TPB, 0, stream>>>(
            pts, batch, bitmap, keyCache, bxyz, ss0, ss1, ss2, sx, sy, sz, n);

        vox_scan_words_kernel<<<nblk, TPB, 0, stream>>>(
            bitmap, wordPrefix, blockSums, nwords);

        vox_scan_blocksums_kernel<<<1, TPB, 0, stream>>>(blockSums, nblk);

        vox_fill_coors_kernel<<<pgrid, TPB, 0, stream>>>(coors, sx, sy, sz, n);

        vox_emit_unique_kernel<<<(K + TPB - 1) / TPB, TPB, 0, stream>>>(
            bitmap, wordPrefix, blockSums, coors, sx, sy, sz, K);

        vox_inv_kernel<<<pgrid, TPB, 0, stream>>>(
            bitmap, wordPrefix, blockSums, keyCache, inv, n);
    }
}